// GATv2Block_35115652612240
// MI455X (gfx1250) — compile-verified
//
#include <hip/hip_runtime.h>
#include <math.h>

#define NN 50000
#define EE 800000
#define IND 256
#define OUTD 128
#define HEADS 4
#define EDGED 64
#define NEG_SLOPE 0.2f
#define LN_EPS 1e-5f
#define SWLD 132   // padded LDS row stride for W_e (bank de-conflict)

typedef __attribute__((ext_vector_type(2))) float v2f;
typedef __attribute__((ext_vector_type(8))) float v8f;

__device__ __forceinline__ unsigned f2ord(float f) {
    unsigned u = __float_as_uint(f);
    return (u & 0x80000000u) ? ~u : (u | 0x80000000u);
}
__device__ __forceinline__ float ord2f(unsigned u) {
    unsigned v = (u & 0x80000000u) ? (u & 0x7fffffffu) : ~u;
    return __uint_as_float(v);
}
__device__ __forceinline__ float wave_sum(float v) {
#pragma unroll
    for (int off = 16; off > 0; off >>= 1) v += __shfl_xor(v, off, 32);
    return v;
}

// ---------------------------------------------------------------------------
// K1: xl = x@W_l + b_l ; xr = x@W_r + b_r  via V_WMMA_F32_16X16X4_F32.
// One wave per 16x16 output tile; 8 waves/block cover all 128 output cols.
// ---------------------------------------------------------------------------
__global__ __launch_bounds__(256) void gemm_xlr_kernel(
    const float* __restrict__ x,
    const float* __restrict__ Wl, const float* __restrict__ bl,
    const float* __restrict__ Wr, const float* __restrict__ br,
    float* __restrict__ xl, float* __restrict__ xr)
{
    const int wave = threadIdx.x >> 5;
    const int lane = threadIdx.x & 31;
    const int m0   = blockIdx.x * 16;
    const int n0   = wave * 16;
    const int half = lane >> 4;
    const int l15  = lane & 15;
    const int koff = half * 2;
    const int ncol = n0 + l15;

    v8f accl = {0.f, 0.f, 0.f, 0.f, 0.f, 0.f, 0.f, 0.f};
    v8f accr = {0.f, 0.f, 0.f, 0.f, 0.f, 0.f, 0.f, 0.f};

    const float* xrow = x + (size_t)(m0 + l15) * IND + koff;

#pragma unroll 4
    for (int k = 0; k < IND; k += 4) {
        v2f a = *(const v2f*)(xrow + k);
        const int r = k + koff;
        v2f b_l2, b_r2;
        b_l2.x = Wl[(size_t)r * OUTD + ncol];
        b_l2.y = Wl[(size_t)(r + 1) * OUTD + ncol];
        b_r2.x = Wr[(size_t)r * OUTD + ncol];
        b_r2.y = Wr[(size_t)(r + 1) * OUTD + ncol];
        accl = __builtin_amdgcn_wmma_f32_16x16x4_f32(
            false, a, false, b_l2, (short)0, accl, false, false);
        accr = __builtin_amdgcn_wmma_f32_16x16x4_f32(
            false, a, false, b_r2, (short)0, accr, false, false);
    }

    const float biasl = bl[ncol];
    const float biasr = br[ncol];
    const int rbase = m0 + half * 8;
#pragma unroll
    for (int i = 0; i < 8; ++i) {
        xl[(size_t)(rbase + i) * OUTD + ncol] = accl[i] + biasl;
        xr[(size_t)(rbase + i) * OUTD + ncol] = accr[i] + biasr;
    }
}

// ---------------------------------------------------------------------------
// K2: degree + segment-sum of edge_attr over dst (thread per edge*channel).
// ---------------------------------------------------------------------------
__global__ void deg_sum_kernel(const int* __restrict__ ei,
                               const float* __restrict__ ea,
                               float* __restrict__ sum_attr,
                               float* __restrict__ deg)
{
    long long t = (long long)blockIdx.x * blockDim.x + threadIdx.x;
    if (t >= (long long)EE * EDGED) return;
    const int e = (int)(t >> 6);
    const int c = (int)(t & 63);
    const int d = ei[EE + e];
    atomicAdd(&sum_attr[(size_t)d * EDGED + c], ea[t]);
    if (c == 0) atomicAdd(&deg[d], 1.0f);
}

// K3: mean_attr = sum_attr / max(deg,1)   (in place)
__global__ void mean_kernel(float* __restrict__ sum_attr,
                            const float* __restrict__ deg)
{
    long long t = (long long)blockIdx.x * blockDim.x + threadIdx.x;
    if (t >= (long long)NN * EDGED) return;
    sum_attr[t] /= fmaxf(deg[t >> 6], 1.0f);
}

// ---------------------------------------------------------------------------
// K4: per-edge GATv2 score + segment max, WMMA edition.
// One wave per 16-edge batch:
//   A (16x64 fp32) = 16 edge-attr rows, 16 v2f fragments per lane.
//   B = W_e tile from LDS; 8 n-tiles x 16 k-steps = 128 WMMAs per wave.
// C layout: for accumulator index i, lanes of a half all hold edge
// (e0 + half*8 + i), channels nt*16 + l15 -> coalesced xl/xr row-segment
// gathers, and per-head score reduces over the 16-lane half.
// ---------------------------------------------------------------------------
__global__ __launch_bounds__(256) void alpha_kernel(
    const int* __restrict__ ei, const float* __restrict__ ea,
    const float* __restrict__ mean_attr, const float* __restrict__ We,
    const float* __restrict__ att,
    const float* __restrict__ xl, const float* __restrict__ xr,
    float* __restrict__ alpha, unsigned* __restrict__ amax)
{
    __shared__ float sW[EDGED * SWLD];   // ~33 KB, padded rows

    const int wave = threadIdx.x >> 5;
    const int lane = threadIdx.x & 31;
    const int half = lane >> 4;
    const int l15  = lane & 15;
    const int koff = half * 2;

    for (int i = threadIdx.x; i < EDGED * OUTD; i += 256) {
        sW[(i >> 7) * SWLD + (i & 127)] = We[i];
    }
    __syncthreads();

    const int e0 = (blockIdx.x * 8 + wave) * 16;
    if (e0 >= (EE + NN)) return;

    // --- A fragments: this lane supplies attr row of edge (e0 + l15) ---
    const int ea_id = e0 + l15;
    const float* arow;
    if (ea_id < EE)            arow = ea + (size_t)ea_id * EDGED;
    else if (ea_id < EE + NN)  arow = mean_attr + (size_t)(ea_id - EE) * EDGED;
    else                       arow = ea;   // dummy (stores guarded below)

    v2f af[16];
#pragma unroll
    for (int kk = 0; kk < 16; ++kk)
        af[kk] = *(const v2f*)(arow + 4 * kk + koff);

    // --- src/dst for the 8 edges owned by this half ---
    int si[8], di[8];
#pragma unroll
    for (int i = 0; i < 8; ++i) {
        const int eg = e0 + half * 8 + i;
        if (eg < EE)           { si[i] = ei[eg]; di[i] = ei[EE + eg]; }
        else if (eg < EE + NN) { si[i] = di[i] = eg - EE; }
        else                   { si[i] = di[i] = 0; }
    }

#pragma unroll
    for (int h = 0; h < HEADS; ++h) {
        float pal[8] = {0.f, 0.f, 0.f, 0.f, 0.f, 0.f, 0.f, 0.f};
#pragma unroll
        for (int sub = 0; sub < 2; ++sub) {
            const int nt = h * 2 + sub;
            const float* sWc = sW + nt * 16 + l15;
            v8f acc = {0.f, 0.f, 0.f, 0.f, 0.f, 0.f, 0.f, 0.f};
#pragma unroll
            for (int kk = 0; kk < 16; ++kk) {
                v2f b;
                b.x = sWc[(4 * kk + koff) * SWLD];
                b.y = sWc[(4 * kk + koff + 1) * SWLD];
                acc = __builtin_amdgcn_wmma_f32_16x16x4_f32(
                    false, af[kk], false, b, (short)0, acc, false, false);
            }
            const int ch = nt * 16 + l15;
            const float attv = att[h * 32 + sub * 16 + l15];
#pragma unroll
            for (int i = 0; i < 8; ++i) {
                const float xlv = xl[(size_t)si[i] * OUTD + ch];
                const float xrv = xr[(size_t)di[i] * OUTD + ch];
                float m = xlv + xrv + acc[i];
                m = m > 0.f ? m : NEG_SLOPE * m;
                pal[i] = fmaf(attv, m, pal[i]);
            }
        }
        // reduce each edge's score over its 16-lane half
#pragma unroll
        for (int i = 0; i < 8; ++i) {
            float r = pal[i];
            r += __shfl_xor(r, 1, 32);
            r += __shfl_xor(r, 2, 32);
            r += __shfl_xor(r, 4, 32);
            r += __shfl_xor(r, 8, 32);
            if (l15 == i) {
                const int eg = e0 + half * 8 + i;
                if (eg < EE + NN) {
                    alpha[(size_t)eg * HEADS + h] = r;
                    atomicMax(&amax[(size_t)di[i] * HEADS + h], f2ord(r));
                }
            }
        }
    }
}

// ---------------------------------------------------------------------------
// K5: ex = exp(alpha - max); denom[dst] += ex; out_acc[dst] += ex * xl[src].
// ---------------------------------------------------------------------------
__global__ __launch_bounds__(256) void agg_kernel(
    const int* __restrict__ ei, const float* __restrict__ xl,
    const float* __restrict__ alpha, const unsigned* __restrict__ amax,
    float* __restrict__ denom, float* __restrict__ out_acc)
{
    const int wave = threadIdx.x >> 5;
    const int lane = threadIdx.x & 31;
    const int e = blockIdx.x * 8 + wave;
    if (e >= (EE + NN)) return;

    int s, d;
    if (e < EE) { s = ei[e]; d = ei[EE + e]; }
    else        { s = d = e - EE; }

    const float4 al = *(const float4*)(alpha + (size_t)e * HEADS);
    const unsigned* mp = amax + (size_t)d * HEADS;
    const float ex0 = expf(al.x - ord2f(mp[0]));
    const float ex1 = expf(al.y - ord2f(mp[1]));
    const float ex2 = expf(al.z - ord2f(mp[2]));
    const float ex3 = expf(al.w - ord2f(mp[3]));

    if (lane == 0) {
        float* dp = denom + (size_t)d * HEADS;
        atomicAdd(&dp[0], ex0);
        atomicAdd(&dp[1], ex1);
        atomicAdd(&dp[2], ex2);
        atomicAdd(&dp[3], ex3);
    }

    const size_t sb = (size_t)s * OUTD + lane;
    float* ob = out_acc + (size_t)d * OUTD + lane;
    atomicAdd(&ob[0],  ex0 * xl[sb]);
    atomicAdd(&ob[32], ex1 * xl[sb + 32]);
    atomicAdd(&ob[64], ex2 * xl[sb + 64]);
    atomicAdd(&ob[96], ex3 * xl[sb + 96]);
}

// ---------------------------------------------------------------------------
// K6: normalize by denom, +bias, LayerNorm over 128 channels, ReLU.
// ---------------------------------------------------------------------------
__global__ __launch_bounds__(256) void final_kernel(
    const float* __restrict__ out_acc, const float* __restrict__ denom,
    const float* __restrict__ bias, const float* __restrict__ gamma,
    const float* __restrict__ beta, float* __restrict__ out)
{
    const int wave = threadIdx.x >> 5;
    const int lane = threadIdx.x & 31;
    const int n = blockIdx.x * 8 + wave;
    if (n >= NN) return;

    const float* ob = out_acc + (size_t)n * OUTD + lane;
    const float* dp = denom + (size_t)n * HEADS;
    float v0 = ob[0]  / dp[0] + bias[lane];
    float v1 = ob[32] / dp[1] + bias[lane + 32];
    float v2 = ob[64] / dp[2] + bias[lane + 64];
    float v3 = ob[96] / dp[3] + bias[lane + 96];

    const float sum = wave_sum(v0 + v1 + v2 + v3);
    const float sq  = wave_sum(v0 * v0 + v1 * v1 + v2 * v2 + v3 * v3);
    const float mu  = sum * (1.0f / OUTD);
    const float var = sq * (1.0f / OUTD) - mu * mu;
    const float rs  = rsqrtf(var + LN_EPS);

    float* op = out + (size_t)n * OUTD + lane;
    float y;
    y = gamma[lane]      * (v0 - mu) * rs + beta[lane];      op[0]  = fmaxf(y, 0.f);
    y = gamma[lane + 32] * (v1 - mu) * rs + beta[lane + 32]; op[32] = fmaxf(y, 0.f);
    y = gamma[lane + 64] * (v2 - mu) * rs + beta[lane + 64]; op[64] = fmaxf(y, 0.f);
    y = gamma[lane + 96] * (v3 - mu) * rs + beta[lane + 96]; op[96] = fmaxf(y, 0.f);
}

extern "C" void kernel_launch(void* const* d_in, const int* in_sizes, int n_in,
                              void* d_out, int out_size, void* d_ws, size_t ws_size,
                              hipStream_t stream)
{
    const float* x     = (const float*)d_in[0];
    const int*   ei    = (const int*)  d_in[1];
    const float* ea    = (const float*)d_in[2];
    const float* Wl    = (const float*)d_in[3];
    const float* bl    = (const float*)d_in[4];
    const float* Wr    = (const float*)d_in[5];
    const float* br    = (const float*)d_in[6];
    const float* We    = (const float*)d_in[7];
    const float* att   = (const float*)d_in[8];
    const float* bias  = (const float*)d_in[9];
    const float* gamma = (const float*)d_in[10];
    const float* beta  = (const float*)d_in[11];
    float* out = (float*)d_out;

    // Workspace layout (floats)
    float* ws = (float*)d_ws;
    size_t off = 0;
    float* xl       = ws + off; off += (size_t)NN * OUTD;          // 25.6 MB
    float* xr       = ws + off; off += (size_t)NN * OUTD;          // 25.6 MB
    float* alpha    = ws + off; off += (size_t)(EE + NN) * HEADS;  // 13.6 MB
    const size_t zoff = off;                                       // zeroed region:
    float*    sum_attr = ws + off;              off += (size_t)NN * EDGED; // -> mean_attr
    float*    deg      = ws + off;              off += (size_t)NN;
    unsigned* amax     = (unsigned*)(ws + off); off += (size_t)NN * HEADS;
    float*    denom    = ws + off;              off += (size_t)NN * HEADS;
    float*    out_acc  = ws + off;              off += (size_t)NN * OUTD;
    const size_t zbytes = (off - zoff) * sizeof(float);

    hipMemsetAsync(ws + zoff, 0, zbytes, stream);

    gemm_xlr_kernel<<<NN / 16, 256, 0, stream>>>(x, Wl, bl, Wr, br, xl, xr);

    {
        long long tot = (long long)EE * EDGED;
        int blocks = (int)((tot + 255) / 256);
        deg_sum_kernel<<<blocks, 256, 0, stream>>>(ei, ea, sum_attr, deg);
    }
    mean_kernel<<<(NN * EDGED + 255) / 256, 256, 0, stream>>>(sum_attr, deg);

    const int etot = EE + NN;
    // alpha: one wave per 16 edges, 8 waves per block
    const int nwaves = (etot + 15) / 16;
    alpha_kernel<<<(nwaves + 7) / 8, 256, 0, stream>>>(ei, ea, sum_attr, We, att,
                                                       xl, xr, alpha, amax);
    agg_kernel<<<(etot + 7) / 8, 256, 0, stream>>>(ei, xl, alpha, amax, denom, out_acc);
    final_kernel<<<(NN + 7) / 8, 256, 0, stream>>>(out_acc, denom, bias, gamma, beta, out);
}